// HashLevel3D_69372311765524
// MI455X (gfx1250) — compile-verified
//
#include <hip/hip_runtime.h>
#include <stdint.h>

// ---------------------------------------------------------------------------
// HashLevel3D (binarized Instant-NGP hash level), MI455X / gfx1250
//
// out[i][f] = sum_c  w_c(i) * sign(table[hash(corner_c(i))][f])
//
// MI455X strategy: the fp32 table only contributes sign bits -> pack it per
// launch into a 2MB bitmask (1 byte/entry, bit f = (feat_f < 0)). 2MB is
// L2/WGP$-resident (192MB L2), turning 512MB of random 32B gathers into
// 16MB of random byte gathers. Streaming traffic (pos in / out) uses NT
// temporal hints so it never evicts the hot packed table. Sign application
// is a rolling-shift XOR on the weight's sign bit (bitop3-friendly).
// ---------------------------------------------------------------------------

#define TABLE_SIZE   2097152u      // 2^21 entries
#define TABLE_MASK   2097151u
#define PRIME_Y      2654435761u
#define PRIME_Z      805459861u
#define RES          64.0f
#define BLK          256

typedef float v4f __attribute__((ext_vector_type(4)));

// ---- pack kernel: 4 table entries (4 x 8 floats) -> one u32 of sign bits ----
__global__ __launch_bounds__(BLK) void pack_signs_kernel(
    const float* __restrict__ table, uint32_t* __restrict__ packed)
{
    uint32_t t = blockIdx.x * BLK + threadIdx.x;          // entry group (4 entries)
    const v4f* tp = (const v4f*)table + (size_t)t * 8u;   // 8 x float4 = 128B/thread
    uint32_t word = 0u;
#pragma unroll
    for (int e = 0; e < 4; ++e) {
        v4f a = tp[2 * e + 0];
        v4f b = tp[2 * e + 1];
        uint32_t byte =
            (a.x < 0.0f ?   1u : 0u) | (a.y < 0.0f ?   2u : 0u) |
            (a.z < 0.0f ?   4u : 0u) | (a.w < 0.0f ?   8u : 0u) |
            (b.x < 0.0f ?  16u : 0u) | (b.y < 0.0f ?  32u : 0u) |
            (b.z < 0.0f ?  64u : 0u) | (b.w < 0.0f ? 128u : 0u);
        word |= byte << (8 * e);
    }
    packed[t] = word;
}

// ---- main kernel: packed byte-gather path ----
__global__ __launch_bounds__(BLK) void hash_interp_packed_kernel(
    const float* __restrict__ pos, const uint8_t* __restrict__ packed,
    float* __restrict__ out, int n)
{
    __shared__ float spos[3 * BLK];
    const int block0 = blockIdx.x * BLK;

    // Prefetch the position stream 2 blocks ahead (global_prefetch_b8).
    if (block0 + 2 * BLK < n)
        __builtin_prefetch(&pos[(size_t)(block0 + 2 * BLK) * 3 + threadIdx.x], 0, 1);

    // Coalesced NT stage of 256 float3 positions through LDS (wave32-friendly).
#pragma unroll
    for (int k = 0; k < 3; ++k) {
        int gi = block0 * 3 + k * BLK + (int)threadIdx.x;
        float v = 0.0f;
        if (gi < n * 3) v = __builtin_nontemporal_load(&pos[gi]);
        spos[k * BLK + threadIdx.x] = v;
    }
    __syncthreads();

    const int i = block0 + (int)threadIdx.x;
    if (i >= n) return;

    float sx = spos[3 * threadIdx.x + 0] * RES;
    float sy = spos[3 * threadIdx.x + 1] * RES;
    float sz = spos[3 * threadIdx.x + 2] * RES;
    float fx = floorf(sx), fy = floorf(sy), fz = floorf(sz);
    float wx = sx - fx,   wy = sy - fy,   wz = sz - fz;

    uint32_t cx = (uint32_t)(int)fx;
    uint32_t cy = (uint32_t)(int)fy;
    uint32_t cz = (uint32_t)(int)fz;

    // hash(x,y,z) low 21 bits == (x*1 ^ y*P1 ^ z*P2) & MASK in u32 arithmetic
    uint32_t hx0 = cx;            uint32_t hx1 = cx + 1u;
    uint32_t hy0 = cy * PRIME_Y;  uint32_t hy1 = hy0 + PRIME_Y;
    uint32_t hz0 = cz * PRIME_Z;  uint32_t hz1 = hz0 + PRIME_Z;

    // 8 independent byte gathers (global_load_u8) into hot 2MB packed table;
    // corner order matches OFFSETS (z fastest, then y, then x).
    uint32_t b0 = packed[(hx0 ^ hy0 ^ hz0) & TABLE_MASK];
    uint32_t b1 = packed[(hx0 ^ hy0 ^ hz1) & TABLE_MASK];
    uint32_t b2 = packed[(hx0 ^ hy1 ^ hz0) & TABLE_MASK];
    uint32_t b3 = packed[(hx0 ^ hy1 ^ hz1) & TABLE_MASK];
    uint32_t b4 = packed[(hx1 ^ hy0 ^ hz0) & TABLE_MASK];
    uint32_t b5 = packed[(hx1 ^ hy0 ^ hz1) & TABLE_MASK];
    uint32_t b6 = packed[(hx1 ^ hy1 ^ hz0) & TABLE_MASK];
    uint32_t b7 = packed[(hx1 ^ hy1 ^ hz1) & TABLE_MASK];

    float ux = 1.0f - wx, uy = 1.0f - wy, uz = 1.0f - wz;
    uint32_t bc[8] = { b0, b1, b2, b3, b4, b5, b6, b7 };
    float    wc[8] = { ux * uy * uz, ux * uy * wz, ux * wy * uz, ux * wy * wz,
                       wx * uy * uz, wx * uy * wz, wx * wy * uz, wx * wy * wz };

    float acc[8] = {0.f, 0.f, 0.f, 0.f, 0.f, 0.f, 0.f, 0.f};
#pragma unroll
    for (int c = 0; c < 8; ++c) {
        // Rolling shift keeps the current feature bit at the sign position;
        // wb ^ (u & 0x80000000) is a single 3-input bit op (v_bitop3-friendly).
        uint32_t u  = bc[c] << 24;            // bit 7 at position 31
        uint32_t wb = __float_as_uint(wc[c]);
#pragma unroll
        for (int f = 7; f >= 0; --f) {
            acc[f] += __uint_as_float(wb ^ (u & 0x80000000u));
            u <<= 1;
        }
    }

    v4f o0 = { acc[0], acc[1], acc[2], acc[3] };
    v4f o1 = { acc[4], acc[5], acc[6], acc[7] };
    v4f* op = (v4f*)(out + (size_t)i * 8);      // 32B-aligned
    __builtin_nontemporal_store(o0, op + 0);    // global_store_b128, TH=NT
    __builtin_nontemporal_store(o1, op + 1);
}

// ---- fallback: direct fp32 gather (only if scratch too small for 2MB) ----
__global__ __launch_bounds__(BLK) void hash_interp_direct_kernel(
    const float* __restrict__ pos, const float* __restrict__ table,
    float* __restrict__ out, int n)
{
    const int i = blockIdx.x * BLK + (int)threadIdx.x;
    if (i >= n) return;

    float sx = pos[(size_t)i * 3 + 0] * RES;
    float sy = pos[(size_t)i * 3 + 1] * RES;
    float sz = pos[(size_t)i * 3 + 2] * RES;
    float fx = floorf(sx), fy = floorf(sy), fz = floorf(sz);
    float wx = sx - fx,   wy = sy - fy,   wz = sz - fz;

    uint32_t cx = (uint32_t)(int)fx, cy = (uint32_t)(int)fy, cz = (uint32_t)(int)fz;
    uint32_t hx0 = cx;            uint32_t hx1 = cx + 1u;
    uint32_t hy0 = cy * PRIME_Y;  uint32_t hy1 = hy0 + PRIME_Y;
    uint32_t hz0 = cz * PRIME_Z;  uint32_t hz1 = hz0 + PRIME_Z;

    uint32_t idx[8] = {
        (hx0 ^ hy0 ^ hz0) & TABLE_MASK, (hx0 ^ hy0 ^ hz1) & TABLE_MASK,
        (hx0 ^ hy1 ^ hz0) & TABLE_MASK, (hx0 ^ hy1 ^ hz1) & TABLE_MASK,
        (hx1 ^ hy0 ^ hz0) & TABLE_MASK, (hx1 ^ hy0 ^ hz1) & TABLE_MASK,
        (hx1 ^ hy1 ^ hz0) & TABLE_MASK, (hx1 ^ hy1 ^ hz1) & TABLE_MASK };

    float ux = 1.0f - wx, uy = 1.0f - wy, uz = 1.0f - wz;
    float wc[8] = { ux * uy * uz, ux * uy * wz, ux * wy * uz, ux * wy * wz,
                    wx * uy * uz, wx * uy * wz, wx * wy * uz, wx * wy * wz };

    float acc[8] = {0.f, 0.f, 0.f, 0.f, 0.f, 0.f, 0.f, 0.f};
#pragma unroll
    for (int c = 0; c < 8; ++c) {
        const v4f* tp = (const v4f*)(table + (size_t)idx[c] * 8);
        v4f a = tp[0], b = tp[1];
        float fv[8] = { a.x, a.y, a.z, a.w, b.x, b.y, b.z, b.w };
        uint32_t wb = __float_as_uint(wc[c]);
#pragma unroll
        for (int f = 0; f < 8; ++f) {
            uint32_t signbit = __float_as_uint(fv[f]) & 0x80000000u;
            acc[f] += __uint_as_float(wb ^ signbit);
        }
    }

    v4f o0 = { acc[0], acc[1], acc[2], acc[3] };
    v4f o1 = { acc[4], acc[5], acc[6], acc[7] };
    v4f* op = (v4f*)(out + (size_t)i * 8);
    __builtin_nontemporal_store(o0, op + 0);
    __builtin_nontemporal_store(o1, op + 1);
}

extern "C" void kernel_launch(void* const* d_in, const int* in_sizes, int n_in,
                              void* d_out, int out_size, void* d_ws, size_t ws_size,
                              hipStream_t stream)
{
    const float* positions = (const float*)d_in[0];   // N x 3 fp32
    const float* table     = (const float*)d_in[1];   // TABLE_SIZE x 8 fp32
    float*       out       = (float*)d_out;           // N x 8 fp32
    const int n = in_sizes[0] / 3;

    if (d_ws != nullptr && ws_size >= (size_t)TABLE_SIZE) {
        uint32_t* packed = (uint32_t*)d_ws;           // 2MB sign-bit table
        pack_signs_kernel<<<TABLE_SIZE / 4u / BLK, BLK, 0, stream>>>(table, packed);
        hash_interp_packed_kernel<<<(n + BLK - 1) / BLK, BLK, 0, stream>>>(
            positions, (const uint8_t*)packed, out, n);
    } else {
        hash_interp_direct_kernel<<<(n + BLK - 1) / BLK, BLK, 0, stream>>>(
            positions, table, out, n);
    }
}